// GraphSAGEWithHidden_32968168964351
// MI455X (gfx1250) — compile-verified
//
#include <hip/hip_runtime.h>

// ---------------------------------------------------------------------------
// GraphSAGE (2x SAGEConv, mean aggr) + log_softmax for MI455X / gfx1250.
//   Layer: h_out = act( mean_agg(h)[dst] @ Wl + h @ Wr + b )
// Edge aggregation is the bandwidth/atomic bound (L2-resident: x/agg/h1 all
// fit in the 192MB L2); the GEMMs (~2.7 GFLOP total) use full-precision
// V_WMMA_F32_16X16X4_F32 so we keep fp32 accuracy at negligible cost.
// ---------------------------------------------------------------------------

typedef __attribute__((ext_vector_type(2))) float v2f;
typedef __attribute__((ext_vector_type(8))) float v8f;

#define CH 64  // IN_CH == HID_CH == 64

// ---------------------------------------------------------------------------
// Edge scatter: agg[dst][c] += feat[src][c]; optional degree count.
// 64 threads per edge -> 256B-coalesced gather and atomic segments.
// ---------------------------------------------------------------------------
__global__ void sage_scatter_kernel(const float* __restrict__ feat,
                                    const int* __restrict__ src,
                                    const int* __restrict__ dst,
                                    float* __restrict__ agg,
                                    float* __restrict__ deg,   // nullptr => skip
                                    int nEdges) {
  long long t = (long long)blockIdx.x * blockDim.x + threadIdx.x;
  int e = (int)(t >> 6);
  int c = (int)(t & 63);
  if (e >= nEdges) return;
  int s = src[e];
  int d = dst[e];
  unsafeAtomicAdd(&agg[(long long)d * CH + c], feat[(long long)s * CH + c]);
  if (deg != nullptr && c == 0) unsafeAtomicAdd(&deg[d], 1.0f);
}

// ---------------------------------------------------------------------------
// Fused SAGE linear: out = act( (agg/max(deg,1)) @ Wl + xin @ Wr + bias )
// One wave computes one 16x16 output tile; K=64 walked in steps of 4 with two
// chained v_wmma_f32_16x16x4_f32 per step (mean@Wl, x@Wr -> same accumulator).
// Fragment layouts per CDNA5 ISA 7.12.2 (32-bit A 16x4, B 4x16, C/D 16x16).
// ---------------------------------------------------------------------------
template <int NOUT, int COLT, int WAVES, bool RELU>
__global__ void sage_linear_wmma_kernel(const float* __restrict__ agg,
                                        const float* __restrict__ deg,
                                        const float* __restrict__ xin,
                                        const float* __restrict__ Wl,
                                        const float* __restrict__ Wr,
                                        const float* __restrict__ bias,
                                        float* __restrict__ out,
                                        int nRows) {
  const int lane = threadIdx.x & 31;
  const int wave = threadIdx.x >> 5;
  const int colTile = wave % COLT;
  const int rowTile = blockIdx.x * (WAVES / COLT) + wave / COLT;
  const int rowBase = rowTile * 16;
  if (rowBase >= nRows) return;  // uniform per wave

  const int rIdx = lane & 15;    // A row within tile / B-C column within tile
  const int half = lane >> 4;    // K sub-pair select (A/B), M+8 select (C/D)
  const int colBase = colTile * 16;
  const int n = colBase + rIdx;
  const bool nok = (n < NOUT);

  const int grow = rowBase + rIdx;
  const float dv = deg[grow];
  const float invd = 1.0f / fmaxf(dv, 1.0f);

  const float* __restrict__ aggRow = agg + (long long)grow * CH;
  const float* __restrict__ xRow  = xin + (long long)grow * CH;

  v8f c = {};  // fp32 accumulator tile

#pragma unroll
  for (int k0 = 0; k0 < CH; k0 += 4) {
    const int k = k0 + 2 * half;  // lanes 0-15: K=k0,k0+1 ; lanes 16-31: K=k0+2,k0+3
    v2f a1, a2, bl, br;
    a1.x = aggRow[k]     * invd;
    a1.y = aggRow[k + 1] * invd;
    a2.x = xRow[k];
    a2.y = xRow[k + 1];
    bl.x = nok ? Wl[k * NOUT + n]       : 0.0f;
    bl.y = nok ? Wl[(k + 1) * NOUT + n] : 0.0f;
    br.x = nok ? Wr[k * NOUT + n]       : 0.0f;
    br.y = nok ? Wr[(k + 1) * NOUT + n] : 0.0f;
    // D = A*B + C  (fp32 WMMA, 8-arg form: neg_a, A, neg_b, B, c_mod, C, reuse_a, reuse_b)
    c = __builtin_amdgcn_wmma_f32_16x16x4_f32(false, a1, false, bl, (short)0, c,
                                              false, false);
    c = __builtin_amdgcn_wmma_f32_16x16x4_f32(false, a2, false, br, (short)0, c,
                                              false, false);
  }

  const float bv = nok ? bias[n] : 0.0f;
#pragma unroll
  for (int r = 0; r < 8; ++r) {
    const int m = rowBase + r + 8 * half;  // C/D: VGPR r holds M=r (lo half), M=r+8 (hi half)
    float v = c[r] + bv;
    if (RELU) v = fmaxf(v, 0.0f);
    if (nok) out[(long long)m * NOUT + n] = v;
  }
}

// ---------------------------------------------------------------------------
// Row-wise log_softmax over 40 logits: one wave32 per row (32 + 8 elements).
// ---------------------------------------------------------------------------
__global__ void log_softmax40_kernel(const float* __restrict__ logits,
                                     float* __restrict__ out, int nRows) {
  const int lane = threadIdx.x & 31;
  const int wave = threadIdx.x >> 5;
  const int row = blockIdx.x * (blockDim.x >> 5) + wave;
  if (row >= nRows) return;
  const float* p = logits + (long long)row * 40;
  float a = p[lane];
  float b = (lane < 8) ? p[32 + lane] : -3.402823466e38f;
  float m = fmaxf(a, b);
#pragma unroll
  for (int off = 16; off > 0; off >>= 1) m = fmaxf(m, __shfl_xor(m, off));
  float s = __expf(a - m) + ((lane < 8) ? __expf(b - m) : 0.0f);
#pragma unroll
  for (int off = 16; off > 0; off >>= 1) s += __shfl_xor(s, off);
  const float lg = m + __logf(s);
  float* o = out + (long long)row * 40;
  o[lane] = a - lg;
  if (lane < 8) o[32 + lane] = b - lg;
}

// ---------------------------------------------------------------------------
extern "C" void kernel_launch(void* const* d_in, const int* in_sizes, int n_in,
                              void* d_out, int out_size, void* d_ws, size_t ws_size,
                              hipStream_t stream) {
  const float* x    = (const float*)d_in[0];
  const int*   ei   = (const int*)d_in[1];   // (2, E): row0=src, row1=dst
  const float* W1l  = (const float*)d_in[2]; // (64,64) row-major
  const float* W1r  = (const float*)d_in[3];
  const float* b1   = (const float*)d_in[4];
  const float* W2l  = (const float*)d_in[5]; // (64,40) row-major
  const float* W2r  = (const float*)d_in[6];
  const float* b2   = (const float*)d_in[7];
  float* outp = (float*)d_out;

  const int nNodes = in_sizes[0] / CH;   // 100000
  const int nEdges = in_sizes[1] / 2;    // 1600000

  // Workspace layout (floats): [deg | agg | h1 | logits]  (~68 MB)
  const long long degCnt = ((long long)nNodes + 255) & ~255LL;
  float* deg    = (float*)d_ws;
  float* agg    = deg + degCnt;
  float* h1     = agg + (long long)nNodes * CH;
  float* logits = h1 + (long long)nNodes * CH;

  const int* src = ei;
  const int* dst = ei + nEdges;

  const long long scatterThreads = (long long)nEdges * CH;
  const int gScatter = (int)((scatterThreads + 255) / 256);
  const int rowTiles = (nNodes + 15) / 16;        // 6250

  // --- Layer 1 ---
  hipMemsetAsync(deg, 0, (size_t)(degCnt + (long long)nNodes * CH) * sizeof(float), stream);
  sage_scatter_kernel<<<gScatter, 256, 0, stream>>>(x, src, dst, agg, deg, nEdges);
  // 8 waves/block: 2 row tiles x 4 col tiles (NOUT=64)
  sage_linear_wmma_kernel<64, 4, 8, true>
      <<<(rowTiles + 1) / 2, 256, 0, stream>>>(agg, deg, x, W1l, W1r, b1, h1, nNodes);

  // --- Layer 2 ---
  hipMemsetAsync(agg, 0, (size_t)((long long)nNodes * CH) * sizeof(float), stream);
  sage_scatter_kernel<<<gScatter, 256, 0, stream>>>(h1, src, dst, agg, nullptr, nEdges);
  // 3 waves/block: 1 row tile x 3 col tiles (NOUT=40, third tile masked)
  sage_linear_wmma_kernel<40, 3, 3, false>
      <<<rowTiles, 96, 0, stream>>>(agg, deg, h1, W2l, W2r, b2, logits, nNodes);

  // --- log_softmax ---
  log_softmax40_kernel<<<(nNodes + 7) / 8, 256, 0, stream>>>(logits, outp, nNodes);
}